// GHMLoss_4818953306440
// MI455X (gfx1250) — compile-verified
//
#include <hip/hip_runtime.h>
#include <math.h>

typedef float v2f __attribute__((ext_vector_type(2)));
typedef float v8f __attribute__((ext_vector_type(8)));

#define BLOCK  256
#define NWAVES (BLOCK / 32)
#define NQ     7
#define MAXB   2048

// Wave32 full-sum via V_WMMA_F32_16X16X4_F32.
// A (16x4): VGPR0 lanes0-15 -> (M=lane,K=0), lanes16-31 -> (M=lane-16,K=2);
// VGPR1 (K=1,K=3) zeroed. B (4x16) = all ones, C = 0.
// => D[m][j] = x[m] + x[m+16] for every column j.
// D layout: VGPR r, lanes0-15 hold D[r][lane], lanes16-31 hold D[r+8][lane-16].
// Sum of one lane's 8 regs covers 8 rows; the xor-16 partner covers the other 8.
__device__ __forceinline__ float wave_sum_wmma(float x) {
    v2f a; a[0] = x;    a[1] = 0.0f;
    v2f b; b[0] = 1.0f; b[1] = 1.0f;
    v8f c = {};
    v8f d = __builtin_amdgcn_wmma_f32_16x16x4_f32(
        /*neg_a=*/false, a, /*neg_b=*/false, b,
        /*c_mod=*/(short)0, c, /*reuse_a=*/false, /*reuse_b=*/false);
    float s = d[0] + d[1] + d[2] + d[3] + d[4] + d[5] + d[6] + d[7];
    s += __shfl_xor(s, 16, 32);
    return s;  // total of all 32 lanes, present in every lane
}

// Fully branchless per-element accumulation: selects only (no t*mask, so a
// non-finite t cannot propagate NaN into a masked-off sum).
__device__ __forceinline__ void accum_elem(float p, float t, float acc[NQ]) {
    // sigmoid via exp(-p): robust at both tails (exp(-p)->inf => sig->0; ->0 => sig->1)
    float en  = __expf(-p);
    float sig = 1.0f / (1.0f + en);
    float g   = fabsf(sig - t);
    bool valid = __builtin_isfinite(g) && __builtin_isfinite(t);

    acc[6] += __expf(p);  // logsumexp over ALL pred (reference log_softmax)

    const float EDGE_LAST = 1.0f + 1e-6f;          // edges[-1] after +1e-6 nudge
    bool  gt  = (t > p);                           // weight = ratio on this side
    float tv  = valid ? t       : 0.0f;            // v_cndmask
    float tpv = valid ? (t * p) : 0.0f;

    acc[0] += valid ? 1.0f : 0.0f;                 // total_samples
    acc[1] += (valid && (g <= EDGE_LAST)) ? 1.0f : 0.0f;  // sum over all bins
    acc[2] += gt ? tv  : 0.0f;                     // A1: sum t    (t >  p)
    acc[3] += gt ? 0.0f : tv;                      // B1: sum t    (t <= p)
    acc[4] += gt ? tpv : 0.0f;                     // A2: sum t*p  (t >  p)
    acc[5] += gt ? 0.0f : tpv;                     // B2: sum t*p  (t <= p)
}

__global__ void __launch_bounds__(BLOCK)
ghm_pass_kernel(const float* __restrict__ pred, const float* __restrict__ tgt,
                float* __restrict__ partials, int n) {
    float acc[NQ];
#pragma unroll
    for (int q = 0; q < NQ; ++q) acc[q] = 0.0f;

    const int nv4    = n >> 2;
    const int stride = gridDim.x * BLOCK;
    const int gtid   = blockIdx.x * BLOCK + threadIdx.x;

    const float4* __restrict__ p4 = (const float4*)pred;
    const float4* __restrict__ t4 = (const float4*)tgt;

    for (int i = gtid; i < nv4; i += stride) {
        float4 p = p4[i];
        float4 t = t4[i];
        accum_elem(p.x, t.x, acc);
        accum_elem(p.y, t.y, acc);
        accum_elem(p.z, t.z, acc);
        accum_elem(p.w, t.w, acc);
    }
    const int tail = n - (nv4 << 2);
    if (gtid < tail) accum_elem(pred[(nv4 << 2) + gtid], tgt[(nv4 << 2) + gtid], acc);

    // wave reduction through the matrix core (uniform flow: EXEC all-ones)
#pragma unroll
    for (int q = 0; q < NQ; ++q) acc[q] = wave_sum_wmma(acc[q]);

    __shared__ float sm[NWAVES][NQ];
    const int wave = threadIdx.x >> 5;
    const int lane = threadIdx.x & 31;
    if (lane == 0) {
#pragma unroll
        for (int q = 0; q < NQ; ++q) sm[wave][q] = acc[q];
    }
    __syncthreads();
    if (threadIdx.x == 0) {
#pragma unroll
        for (int q = 0; q < NQ; ++q) {
            float s = 0.0f;
            for (int w = 0; w < NWAVES; ++w) s += sm[w][q];
            partials[blockIdx.x * NQ + q] = s;
        }
    }
}

__global__ void __launch_bounds__(BLOCK)
ghm_final_kernel(const float* __restrict__ partials, int nblocks,
                 float* __restrict__ out) {
    float acc[NQ];
#pragma unroll
    for (int q = 0; q < NQ; ++q) acc[q] = 0.0f;

    for (int b = threadIdx.x; b < nblocks; b += BLOCK) {
#pragma unroll
        for (int q = 0; q < NQ; ++q) acc[q] += partials[b * NQ + q];
    }

#pragma unroll
    for (int q = 0; q < NQ; ++q) acc[q] = wave_sum_wmma(acc[q]);

    __shared__ float sm[NWAVES][NQ];
    const int wave = threadIdx.x >> 5;
    const int lane = threadIdx.x & 31;
    if (lane == 0) {
#pragma unroll
        for (int q = 0; q < NQ; ++q) sm[wave][q] = acc[q];
    }
    __syncthreads();
    if (threadIdx.x == 0) {
        float tot[NQ];
#pragma unroll
        for (int q = 0; q < NQ; ++q) {
            float s = 0.0f;
            for (int w = 0; w < NWAVES; ++w) s += sm[w][q];
            tot[q] = s;
        }
        float total_samples   = tot[0];
        float total_positives = 0.5f * tot[1];            // momentum EMA of zero state
        float total_negatives = total_samples - total_positives;  // pre-clamp
        float tp    = fmaxf(total_positives, 1.0f);
        float ratio = total_negatives / tp;

        float Swt  = ratio * tot[2] + tot[3];  // sum w*t
        float Swtp = ratio * tot[4] + tot[5];  // sum w*t*pred
        float lse  = logf(tot[6]);             // log sum exp(pred)

        out[0] = lse * Swt - Swtp;             // -sum w*t*(pred - lse)
    }
}

extern "C" void kernel_launch(void* const* d_in, const int* in_sizes, int n_in,
                              void* d_out, int out_size, void* d_ws, size_t ws_size,
                              hipStream_t stream) {
    const float* pred = (const float*)d_in[0];
    const float* tgt  = (const float*)d_in[1];
    float* out        = (float*)d_out;
    float* partials   = (float*)d_ws;  // MAXB * NQ floats = 56 KB max
    const int n       = in_sizes[0];

    int nv4 = n >> 2;
    int nb  = (nv4 + BLOCK - 1) / BLOCK;
    if (nb > MAXB) nb = MAXB;
    if (nb < 1)    nb = 1;

    ghm_pass_kernel<<<nb, BLOCK, 0, stream>>>(pred, tgt, partials, n);
    ghm_final_kernel<<<1, BLOCK, 0, stream>>>(partials, nb, out);
}